// GraphConvBlock_11227044512390
// MI455X (gfx1250) — compile-verified
//
#include <hip/hip_runtime.h>
#include <hip/hip_bf16.h>

// Problem constants (match reference)
#define HGT 256
#define WID 256
#define CH  64
#define NB  2
#define HW  (HGT * WID)     // 65536 nodes per image
#define NN  (NB * HW)       // 131072 total nodes
#define NEG_SLOPE 0.2f

typedef __attribute__((ext_vector_type(2))) float v2f;
typedef __attribute__((ext_vector_type(8))) float v8f;

#define LDA 66  // padded LDS stride for lin_w^T (even -> 8B aligned float2, spreads banks)

// ---------------------------------------------------------------------------
// Kernel 1: hfeat_t[b][co][hw] = sum_ci lin_w[ci][co] * (x[b][ci][hw] + pe(ci,hw))
// computed with V_WMMA_F32_16X16X4_F32. Also produces per-node attention
// scalars a_src / a_dst by reducing the full 64-row output column each wave
// already holds in its accumulators.
//
// Wave mapping: tile = 16 consecutive nodes (columns). 8 waves / block.
// A = lin_w^T (16x4 frags from LDS), B = xg_t (16 cols x 4 k), C/D = v8f.
// Per-lane fragment layout per ISA: M/N = lane%16, K-pair = k0 + (lane>>4)*2.
// ---------------------------------------------------------------------------
__global__ __launch_bounds__(256) void gat_gemm_kernel(
    const float* __restrict__ x, const float* __restrict__ pos_w,
    const float* __restrict__ pos_b, const float* __restrict__ lin_w,
    const float* __restrict__ att_src, const float* __restrict__ att_dst,
    float* __restrict__ hfeat, float* __restrict__ a_src,
    float* __restrict__ a_dst)
{
    __shared__ float A_lds[CH * LDA];   // lin_w^T: A_lds[co*LDA + ci]
    __shared__ float pw0[CH], pw1[CH], pbv[CH], asl[CH], adl[CH];

    const int tid = threadIdx.x;

    // Stage lin_w transposed into LDS (coalesced global reads).
    for (int i = tid; i < CH * CH; i += 256) {
        int ci = i >> 6, co = i & 63;            // lin_w[ci][co]
        A_lds[co * LDA + ci] = lin_w[i];
    }
    if (tid < CH) {
        pw0[tid] = pos_w[tid];
        pw1[tid] = pos_w[CH + tid];
        pbv[tid] = pos_b[tid];
        asl[tid] = att_src[tid];
        adl[tid] = att_dst[tid];
    }
    __syncthreads();

    const int tile = blockIdx.x * 8 + (tid >> 5);  // 8192 tiles total
    const int b    = tile >> 12;                   // 4096 tiles per image
    const int n0   = (tile & 4095) << 4;           // first node column of tile

    const int lane = tid & 31;
    const int half = lane >> 4;
    const int l16  = lane & 15;

    const int n  = n0 + l16;        // node (hw) this lane owns for B/D
    const int hh = n >> 8;
    const int ww = n & 255;
    const float py = -1.f + (2.f / 255.f) * (float)hh;
    const float px = -1.f + (2.f / 255.f) * (float)ww;

    const float* xb = x + (size_t)b * CH * HW;

    v8f acc[4] = {};   // 4 M-tiles of 16 rows -> all 64 output channels

    #pragma unroll
    for (int k0 = 0; k0 < CH; k0 += 4) {
        const int kb = k0 + half * 2;   // this lane's K pair: kb, kb+1

        // B fragment: xg_t[kb..kb+1][n] = x + positional encoding (fused)
        const float xa = xb[(size_t)kb       * HW + n];
        const float xc = xb[(size_t)(kb + 1) * HW + n];
        const v2f w0 = *(const v2f*)&pw0[kb];
        const v2f w1 = *(const v2f*)&pw1[kb];
        const v2f bb = *(const v2f*)&pbv[kb];
        v2f bf;
        bf.x = xa + py * w0.x + px * w1.x + bb.x;
        bf.y = xc + py * w0.y + px * w1.y + bb.y;

        #pragma unroll
        for (int mt = 0; mt < 4; ++mt) {
            // A fragment: lin_w^T[m][kb..kb+1] from LDS (8B aligned)
            const v2f af = *(const v2f*)&A_lds[(mt * 16 + l16) * LDA + kb];
            acc[mt] = __builtin_amdgcn_wmma_f32_16x16x4_f32(
                false, af, false, bf, (short)0, acc[mt], false, false);
        }
    }

    // Store hfeat_t (channel-major) + reduce attention scalars.
    // D layout: VGPR r -> M = mt*16 + half*8 + r, N = lane%16.
    float* hf = hfeat + (size_t)b * CH * HW;
    float ssum = 0.f, dsum = 0.f;
    #pragma unroll
    for (int mt = 0; mt < 4; ++mt) {
        #pragma unroll
        for (int r = 0; r < 8; ++r) {
            const int co = mt * 16 + half * 8 + r;
            const float v = acc[mt][r];
            hf[(size_t)co * HW + n] = v;
            ssum += v * asl[co];
            dsum += v * adl[co];
        }
    }
    // lanes l and l+16 hold partials for the same node n -> combine halves
    ssum += __shfl_xor(ssum, 16, 32);
    dsum += __shfl_xor(dsum, 16, 32);
    if (half == 0) {
        a_src[b * HW + n] = ssum;
        a_dst[b * HW + n] = dsum;
    }
}

// ---------------------------------------------------------------------------
// Kernel 2: per-node softmax over the <=9 incoming stencil edges.
// alpha stored SoA: alpha[j * NN + node]  (coalesced loads in kernel 3).
// ---------------------------------------------------------------------------
__global__ __launch_bounds__(256) void gat_alpha_kernel(
    const float* __restrict__ a_src, const float* __restrict__ a_dst,
    float* __restrict__ alpha)
{
    const int gid = blockIdx.x * 256 + threadIdx.x;
    if (gid >= NN) return;

    const int b  = gid >> 16;
    const int hw = gid & 65535;
    const int h  = hw >> 8;
    const int w  = hw & 255;

    const float ad = a_dst[gid];

    float e[9];
    float m = -3.4e38f;
    #pragma unroll
    for (int j = 0; j < 9; ++j) {
        const int dr = j / 3 - 1;
        const int dc = j % 3 - 1;
        const int hs = h + dr;
        const int ws = w + dc;
        float v = -3.4e38f;
        if ((unsigned)hs < (unsigned)HGT && (unsigned)ws < (unsigned)WID) {
            const float t = a_src[(b << 16) + (hs << 8) + ws] + ad;
            v = (t > 0.f) ? t : NEG_SLOPE * t;   // leaky_relu
        }
        e[j] = v;
        m = fmaxf(m, v);
    }

    float p[9];
    float s = 0.f;
    #pragma unroll
    for (int j = 0; j < 9; ++j) {
        p[j] = (e[j] > -3.0e38f) ? __expf(e[j] - m) : 0.f;
        s += p[j];
    }
    const float inv = 1.f / s;   // self-loop always present -> s > 0
    #pragma unroll
    for (int j = 0; j < 9; ++j)
        alpha[(size_t)j * NN + gid] = p[j] * inv;
}

// ---------------------------------------------------------------------------
// Kernel 3: out[b][c][h][w] = bias[c] + sum_j alpha[j][node] *
//                             hfeat_t[b][c][h+dr][w+dc]
// Channel-major everywhere -> shifted-row loads & stores are coalesced.
// ---------------------------------------------------------------------------
__global__ __launch_bounds__(256) void gat_agg_kernel(
    const float* __restrict__ hfeat, const float* __restrict__ alpha,
    const float* __restrict__ bias, float* __restrict__ out)
{
    const int gid = blockIdx.x * 256 + threadIdx.x;   // (b,c,h,w) flattened
    if (gid >= NB * CH * HW) return;

    const int hw = gid & 65535;
    const int h  = hw >> 8;
    const int w  = hw & 255;
    const int c  = (gid >> 16) & 63;
    const int b  = gid >> 22;
    const int node = (b << 16) | hw;

    const float* hrow = hfeat + ((size_t)gid - hw);  // base of (b,c) plane

    float acc = bias[c];
    #pragma unroll
    for (int j = 0; j < 9; ++j) {
        const int dr = j / 3 - 1;
        const int dc = j % 3 - 1;
        const int hs = h + dr;
        const int ws = w + dc;
        if ((unsigned)hs < (unsigned)HGT && (unsigned)ws < (unsigned)WID) {
            acc += alpha[(size_t)j * NN + node] * hrow[(hs << 8) + ws];
        }
    }
    out[gid] = acc;
}

// ---------------------------------------------------------------------------
extern "C" void kernel_launch(void* const* d_in, const int* in_sizes, int n_in,
                              void* d_out, int out_size, void* d_ws, size_t ws_size,
                              hipStream_t stream)
{
    const float* x       = (const float*)d_in[0];
    const float* pos_w   = (const float*)d_in[1];
    const float* pos_b   = (const float*)d_in[2];
    const float* lin_w   = (const float*)d_in[3];
    const float* att_src = (const float*)d_in[4];
    const float* att_dst = (const float*)d_in[5];
    const float* bias    = (const float*)d_in[6];
    // d_in[7]/d_in[8] (edge_src/edge_dst) unused: the graph is the known
    // 3x3 grid stencil with self-loops; we exploit the structure directly.
    (void)in_sizes; (void)n_in; (void)out_size; (void)ws_size;

    float* ws    = (float*)d_ws;
    float* hfeat = ws;                                // NB*CH*HW = 8,388,608 f
    float* a_src = hfeat + (size_t)NB * CH * HW;      // NN
    float* a_dst = a_src + NN;                        // NN
    float* alpha = a_dst + NN;                        // 9*NN

    // K1: 8192 wave-tiles / 8 waves per block = 1024 blocks
    gat_gemm_kernel<<<1024, 256, 0, stream>>>(
        x, pos_w, pos_b, lin_w, att_src, att_dst, hfeat, a_src, a_dst);

    // K2: one thread per node
    gat_alpha_kernel<<<NN / 256, 256, 0, stream>>>(a_src, a_dst, alpha);

    // K3: one thread per output element
    gat_agg_kernel<<<(NB * CH * HW) / 256, 256, 0, stream>>>(
        hfeat, alpha, bias, (float*)d_out);
}